// VoLUNet_936302870625
// MI455X (gfx1250) — compile-verified
//
#include <hip/hip_runtime.h>
#include <math.h>

// ---------------- problem constants ----------------
#define HW   1048576
#define C    64
#define KSEG 4096
#define H1   256
#define NCLS 16
// B-spline: G=5, SO=3 -> 12 uniform knots t_j = 0.4*(j-3)-1, 8 cubic bases
#define KT1  (C + C*8)        // 576  augmented K for layer 1
#define KT2  (H1 + H1*8)      // 2304 augmented K for layer 2

typedef __attribute__((ext_vector_type(2))) float v2f;
typedef __attribute__((ext_vector_type(4))) float v4f;
typedef __attribute__((ext_vector_type(8))) float v8f;

// ---------------- device helpers ----------------
__device__ __forceinline__ float silu_f(float x) {
    return x / (1.0f + expf(-x));
}

// cubic B-spline bases on the fixed uniform grid (matches reference _b_splines)
__device__ __forceinline__ void bspline8(float x, float* __restrict__ out) {
    float t[12];
#pragma unroll
    for (int j = 0; j < 12; ++j) t[j] = 0.4f * (float)(j - 3) - 1.0f;
    float b[11];
#pragma unroll
    for (int j = 0; j < 11; ++j) b[j] = (x >= t[j] && x < t[j + 1]) ? 1.0f : 0.0f;
#pragma unroll
    for (int k = 1; k <= 3; ++k) {
#pragma unroll
        for (int j = 0; j < 10; ++j) {
            if (j < 11 - k) {
                float left  = (x - t[j])         / (t[j + k]     - t[j]);
                float right = (t[j + k + 1] - x) / (t[j + k + 1] - t[j + 1]);
                b[j] = left * b[j] + right * b[j + 1];
            }
        }
    }
#pragma unroll
    for (int g = 0; g < 8; ++g) out[g] = b[g];
}

// ---------------- kernels ----------------
__global__ void zero_kernel(float* __restrict__ p, int n) {
    int i = blockIdx.x * blockDim.x + threadIdx.x;
    if (i < n) p[i] = 0.0f;
}

// segment sum: acc[seg][c] += x[row][c], cnt[seg] += 1
// 16 threads per row, float4 per thread -> 64 cols; 16 rows per 256-thread block iter.
// x / cls are streamed read-once -> non-temporal loads (TH=NT) so the 1MB atomic
// accumulator stays resident in L2 while 256MB streams past.
__global__ void segsum_kernel(const float* __restrict__ x, const int* __restrict__ cls,
                              float* __restrict__ acc, float* __restrict__ cnt) {
    int t = threadIdx.x;
    int rloc = t >> 4;          // 0..15
    int c4   = (t & 15) << 2;   // 0,4,...,60
    for (int base = blockIdx.x * 16; base < HW; base += gridDim.x * 16) {
        int row = base + rloc;
        int seg = __builtin_nontemporal_load(cls + row);
        const v4f v = __builtin_nontemporal_load((const v4f*)(x + (size_t)row * C + c4));
        float* a = acc + (size_t)seg * C + c4;
        atomicAdd(a + 0, v.x);
        atomicAdd(a + 1, v.y);
        atomicAdd(a + 2, v.z);
        atomicAdd(a + 3, v.w);
        if ((t & 15) == 0) atomicAdd(cnt + seg, 1.0f);
    }
}

// build transposed combined weight for layer 1: Wt1[k][o], k in [0,576), o in [0,256)
__global__ void wprep1_kernel(const float* __restrict__ bw1, const float* __restrict__ sw1,
                              const float* __restrict__ sc1, float* __restrict__ Wt1) {
    int idx = blockIdx.x * blockDim.x + threadIdx.x;      // o*576 + j
    if (idx >= H1 * KT1) return;
    int o = idx / KT1, j = idx % KT1;
    float v;
    if (j < C) {
        v = bw1[o * C + j];
    } else {
        int j2 = j - C, i = j2 >> 3, g = j2 & 7;
        v = sw1[((size_t)o * C + i) * 8 + g] * sc1[o * C + i];
    }
    Wt1[(size_t)j * H1 + o] = v;
}

// build transposed combined weight for layer 2: Wt2[k][o], k in [0,2304), o in [0,16)
__global__ void wprep2_kernel(const float* __restrict__ bw2, const float* __restrict__ sw2,
                              const float* __restrict__ sc2, float* __restrict__ Wt2) {
    int idx = blockIdx.x * blockDim.x + threadIdx.x;      // o*2304 + j
    if (idx >= NCLS * KT2) return;
    int o = idx / KT2, j = idx % KT2;
    float v;
    if (j < H1) {
        v = bw2[o * H1 + j];
    } else {
        int j2 = j - H1, i = j2 >> 3, g = j2 & 7;
        v = sw2[((size_t)o * H1 + i) * 8 + g] * sc2[o * H1 + i];
    }
    Wt2[(size_t)j * NCLS + o] = v;
}

// cluster mean + augmented activation for layer 1: A1[k][0:64]=silu(f), A1[k][64+c*8+g]=bases
__global__ void aprep1_kernel(const float* __restrict__ acc, const float* __restrict__ cnt,
                              float* __restrict__ A1) {
    int idx = blockIdx.x * blockDim.x + threadIdx.x;      // k*64 + c
    if (idx >= KSEG * C) return;
    int k = idx >> 6, c = idx & 63;
    float n = cnt[k];
    float f = acc[(size_t)k * C + c] / fmaxf(n, 1.0f);
    float* row = A1 + (size_t)k * KT1;
    row[c] = silu_f(f);
    float b[8];
    bspline8(f, b);
#pragma unroll
    for (int g = 0; g < 8; ++g) row[C + c * 8 + g] = b[g];
}

// augmented activation for layer 2 from h (4096x256)
__global__ void aprep2_kernel(const float* __restrict__ h, float* __restrict__ A2) {
    int idx = blockIdx.x * blockDim.x + threadIdx.x;      // m*256 + i
    if (idx >= KSEG * H1) return;
    int m = idx >> 8, i = idx & 255;
    float f = h[(size_t)m * H1 + i];
    float* row = A2 + (size_t)m * KT2;
    row[i] = silu_f(f);
    float b[8];
    bspline8(f, b);
#pragma unroll
    for (int g = 0; g < 8; ++g) row[H1 + i * 8 + g] = b[g];
}

// f32 WMMA GEMM: D[M][N] = A[M][Kt] * Bt[Kt][N]   (Bt is k-major = W transposed)
// one wave per 16-row strip x (16*NT)-col strip; v_wmma_f32_16x16x4_f32 per 4-k step.
// A fragment (16x4 f32): lane<16 -> M=lane, K=kk+{0,1}; lane>=16 -> M=lane-16, K=kk+{2,3}
// B fragment (4x16 f32): symmetric; C/D: vgpr r, lane L -> row r+8*(L>>4), col L&15.
template <int NT>
__global__ void gemm_f32_wmma(const float* __restrict__ A, const float* __restrict__ Bt,
                              float* __restrict__ D, int M, int Kt, int N) {
    const int tilesN = N / (16 * NT);
    const int tm = blockIdx.x / tilesN;
    const int tn = blockIdx.x % tilesN;
    const int lane = threadIdx.x & 31;
    const int mrow = lane & 15;
    const int khalf = (lane >> 4) << 1;   // 0 or 2
    const int n0 = tn * 16 * NT;

    const float* Arow = A + (size_t)(tm * 16 + mrow) * Kt;

    v8f accv[NT];
#pragma unroll
    for (int t = 0; t < NT; ++t) accv[t] = (v8f){0.f, 0.f, 0.f, 0.f, 0.f, 0.f, 0.f, 0.f};

    for (int kk = 0; kk < Kt; kk += 4) {
        v2f a;
        a.x = Arow[kk + khalf];
        a.y = Arow[kk + khalf + 1];
#pragma unroll
        for (int t = 0; t < NT; ++t) {
            int n = n0 + t * 16 + mrow;
            v2f b;
            b.x = Bt[(size_t)(kk + khalf) * N + n];
            b.y = Bt[(size_t)(kk + khalf + 1) * N + n];
            accv[t] = __builtin_amdgcn_wmma_f32_16x16x4_f32(
                false, a, false, b, (short)0, accv[t], false, false);
        }
    }

    const int rbase = (lane >> 4) * 8;
#pragma unroll
    for (int t = 0; t < NT; ++t) {
#pragma unroll
        for (int r = 0; r < 8; ++r) {
            D[(size_t)(tm * 16 + rbase + r) * N + n0 + t * 16 + mrow] = accv[t][r];
        }
    }
}

// ---------------- host launcher ----------------
extern "C" void kernel_launch(void* const* d_in, const int* in_sizes, int n_in,
                              void* d_out, int out_size, void* d_ws, size_t ws_size,
                              hipStream_t stream) {
    const float* x   = (const float*)d_in[0];
    const int*   cls = (const int*)  d_in[1];
    const float* bw1 = (const float*)d_in[2];
    const float* sw1 = (const float*)d_in[3];
    const float* sc1 = (const float*)d_in[4];
    const float* bw2 = (const float*)d_in[5];
    const float* sw2 = (const float*)d_in[6];
    const float* sc2 = (const float*)d_in[7];
    float* out = (float*)d_out;

    float* ws  = (float*)d_ws;
    float* acc = ws;                        // KSEG*C
    float* cnt = acc + (size_t)KSEG * C;    // KSEG
    float* A1  = cnt + KSEG;                // KSEG*KT1
    float* Wt1 = A1 + (size_t)KSEG * KT1;   // KT1*H1
    float* h   = Wt1 + (size_t)KT1 * H1;    // KSEG*H1
    float* A2  = h + (size_t)KSEG * H1;     // KSEG*KT2
    float* Wt2 = A2 + (size_t)KSEG * KT2;   // KT2*NCLS

    // 1. zero atomic accumulators (every call: atomics must start clean)
    {
        int n = KSEG * C + KSEG;
        zero_kernel<<<(n + 255) / 256, 256, 0, stream>>>(acc, n);
    }
    // 2. bandwidth phase: segment sums via global f32 atomics (acc fits in L2)
    segsum_kernel<<<8192, 256, 0, stream>>>(x, cls, acc, cnt);
    // 3. layer-1 combined weight (transposed, k-major)
    {
        int n = H1 * KT1;
        wprep1_kernel<<<(n + 255) / 256, 256, 0, stream>>>(bw1, sw1, sc1, Wt1);
    }
    // 4. cluster means + silu/spline augmentation
    {
        int n = KSEG * C;
        aprep1_kernel<<<(n + 255) / 256, 256, 0, stream>>>(acc, cnt, A1);
    }
    // 5. layer-1 GEMM: h = A1(4096x576) * Wt1(576x256), 4 n-tiles/wave
    gemm_f32_wmma<4><<<(KSEG / 16) * (H1 / 64), 32, 0, stream>>>(A1, Wt1, h, KSEG, KT1, H1);
    // 6. layer-2 augmentation
    {
        int n = KSEG * H1;
        aprep2_kernel<<<(n + 255) / 256, 256, 0, stream>>>(h, A2);
    }
    // 7. layer-2 combined weight
    {
        int n = NCLS * KT2;
        wprep2_kernel<<<(n + 255) / 256, 256, 0, stream>>>(bw2, sw2, sc2, Wt2);
    }
    // 8. layer-2 GEMM -> out (4096x16)
    gemm_f32_wmma<1><<<(KSEG / 16) * (NCLS / 16), 32, 0, stream>>>(A2, Wt2, out, KSEG, KT2, NCLS);
}